// TriangleTensorNetwork_14259291422905
// MI455X (gfx1250) — compile-verified
//
#include <hip/hip_runtime.h>

// ---------------------------------------------------------------------------
// TriangleTensorNetwork on MI455X (gfx1250), full f32 via V_WMMA_F32_16X16X4.
//
// Reformulation:  out[b] = sum_{s,r} p1[b,s] * p2[b,r] * Q[b,s,r]
//                 Q[b,s,r] = sum_o p3[b,o] * T[s,r,o]
//                 T[s,r,o] = sum_{x,y,z} cube_s[s,y,z] cube_r[x,r,z] cube_o[x,y,o]
// T is batch-independent (64^3 = 1 MB, ~4.3 GFLOP once) -> every stage is a
// shared-weight GEMM over the batch, ideal for per-wave WMMA tiles.
//
// LDS layouts:
//   A-side tiles  : row-major, row stride LSTR=68 floats (bank-spread, rows
//                   16B aligned; (k,k+1) pair is contiguous -> ds_load_b64).
//   B-side tiles  : pair-interleaved [k/2][n][2], pair-row stride PSTR=160
//                   floats. A lane's (k,k+1) operand pair is one 8B-aligned
//                   ds_load_b64 straight into an even VGPR pair (no v_mov
//                   repair), and 160 mod 64 == 32 puts the hi-half lanes 32
//                   banks away from the lo-half -> all 64 banks, no conflict.
// ---------------------------------------------------------------------------

typedef __attribute__((ext_vector_type(2))) float v2f;
typedef __attribute__((ext_vector_type(8))) float v8f;

#define LSTR 68   // A-side padded row stride (floats)
#define PSTR 160  // B-side pair-row stride (floats); 160 % 64 == 32

__device__ __forceinline__ v8f wmma_f32(v2f a, v2f b, v8f c) {
  // D(16x16,f32) = A(16x4,f32) * B(4x16,f32) + C   -- v_wmma_f32_16x16x4_f32
  return __builtin_amdgcn_wmma_f32_16x16x4_f32(
      /*neg_a=*/false, a, /*neg_b=*/false, b,
      /*c_mod=*/(short)0, c, /*reuse_a=*/false, /*reuse_b=*/false);
}

__device__ __forceinline__ v8f v8f_zero() {
  v8f z;
#pragma unroll
  for (int i = 0; i < 8; ++i) z[i] = 0.0f;
  return z;
}

__device__ __forceinline__ v2f ld_v2f(const float* p) {
  return *reinterpret_cast<const v2f*>(p);  // 8B-aligned by construction
}

// ---------------------------------------------------------------------------
// Y[B,64] = act(X[B,Ktot] @ W + bias)
//   transW=0: W stored [Ktot,64] (proj_s / proj_o / proj_r)
//   transW=1: W stored [64,Ktot] and used transposed (W21/W22/W23, i.e. X@W^T)
// Workgroup: 64 batch rows, 4 waves x 16 rows, 4 column tiles each.
// ---------------------------------------------------------------------------
__global__ __launch_bounds__(128) void fused_gemm_bx64(
    const float* __restrict__ X, const float* __restrict__ W,
    const float* __restrict__ bias, float* __restrict__ Y,
    int Ktot, int transW, int doRelu) {
  __shared__ float sX[64 * LSTR];    // A tile: [row][k]
  __shared__ float sWp[32 * PSTR];   // B tile: [k/2][n][2]
  const int tid = threadIdx.x;
  const int wave = tid >> 5;
  const int lane = tid & 31;
  const int lm = lane & 15;  // row (A) / col (B,C) within 16x16 tile
  const int hi = lane >> 4;  // K-pair select for A/B, +8 row select for C/D
  const int rowBase = blockIdx.x * 64;

  v8f acc[4];
#pragma unroll
  for (int nt = 0; nt < 4; ++nt) acc[nt] = v8f_zero();

  for (int kk = 0; kk < Ktot; kk += 64) {
    __syncthreads();
    // Stage X chunk [64 rows x 64 k], coalesced float4 along k.
#pragma unroll
    for (int it = 0; it < 8; ++it) {
      const int idx = tid + it * 128;  // 1024 float4 slots
      const int r = idx >> 4, c4 = idx & 15;
      const float4 v = *reinterpret_cast<const float4*>(
          &X[(size_t)(rowBase + r) * Ktot + kk + c4 * 4]);
      *reinterpret_cast<float4*>(&sX[r * LSTR + c4 * 4]) = v;
    }
    // Stage W chunk into pair-interleaved [k/2][n][2] layout.
    if (transW) {
#pragma unroll
      for (int it = 0; it < 8; ++it) {
        const int idx = tid + it * 128;
        const int n = idx >> 4, c4 = idx & 15;  // 4 consecutive k for col n
        const float4 v = *reinterpret_cast<const float4*>(
            &W[(size_t)n * Ktot + kk + c4 * 4]);
        const int pr = (c4 * 4) >> 1;  // even k0 -> clean pair rows
        sWp[(pr + 0) * PSTR + n * 2 + 0] = v.x;
        sWp[(pr + 0) * PSTR + n * 2 + 1] = v.y;
        sWp[(pr + 1) * PSTR + n * 2 + 0] = v.z;
        sWp[(pr + 1) * PSTR + n * 2 + 1] = v.w;
      }
    } else {
#pragma unroll
      for (int it = 0; it < 8; ++it) {
        const int idx = tid + it * 128;
        const int k = idx >> 4, c4 = idx & 15;  // 4 consecutive n for row k
        const float4 v = *reinterpret_cast<const float4*>(
            &W[(size_t)(kk + k) * 64 + c4 * 4]);
        float* dst = &sWp[(k >> 1) * PSTR + (k & 1)];
        dst[(c4 * 4 + 0) * 2] = v.x;
        dst[(c4 * 4 + 1) * 2] = v.y;
        dst[(c4 * 4 + 2) * 2] = v.z;
        dst[(c4 * 4 + 3) * 2] = v.w;
      }
    }
    __syncthreads();
    // 16 K-steps of WMMA (K=64 per chunk, 4 per instruction).
    const int arow = wave * 16 + lm;
#pragma unroll
    for (int st = 0; st < 16; ++st) {
      const int kb = st * 4 + hi * 2;
      const v2f a = ld_v2f(&sX[arow * LSTR + kb]);
      const int bpr = st * 2 + hi;  // pair-row = kb>>1
#pragma unroll
      for (int nt = 0; nt < 4; ++nt) {
        const v2f b = ld_v2f(&sWp[bpr * PSTR + (nt * 16 + lm) * 2]);
        acc[nt] = wmma_f32(a, b, acc[nt]);
      }
    }
  }
  // Epilogue: bias + ReLU + store (C/D layout: VGPR j -> row j + hi*8).
#pragma unroll
  for (int nt = 0; nt < 4; ++nt) {
    const int col = nt * 16 + lm;
    const float bv = bias ? bias[col] : 0.0f;
#pragma unroll
    for (int j = 0; j < 8; ++j) {
      const int r = wave * 16 + j + hi * 8;
      float v = acc[nt][j] + bv;
      if (doRelu) v = fmaxf(v, 0.0f);
      Y[(size_t)(rowBase + r) * 64 + col] = v;
    }
  }
}

// ---------------------------------------------------------------------------
// T[s,r,o] = sum_x ( V_x^T @ Co_x ),  V_x[y,r] = sum_z Cs_s[y,z] * Cr_x[r,z]
// One workgroup per s; T_s[64r x 64o] accumulates in registers across x.
// ---------------------------------------------------------------------------
__global__ __launch_bounds__(128) void tcube_kernel(
    const float* __restrict__ cube_s, const float* __restrict__ cube_r,
    const float* __restrict__ cube_o, float* __restrict__ T) {
  __shared__ float sCs[64 * LSTR];    // A (step1): [y][z]
  __shared__ float sCr[64 * LSTR];    // B (step1): [r][z]; (z,z+1) contiguous
  __shared__ float sCoP[32 * PSTR];   // B (step2): [y/2][o][2]
  __shared__ float sVP[32 * PSTR];    // A (step2): [y/2][r][2]
  const int tid = threadIdx.x, wave = tid >> 5, lane = tid & 31;
  const int lm = lane & 15, hi = lane >> 4;
  const int s = blockIdx.x;

  // Stage Cs_s once (16 KB, reused for all 64 x-iterations).
#pragma unroll
  for (int it = 0; it < 8; ++it) {
    const int idx = tid + it * 128;
    const int r = idx >> 4, c4 = idx & 15;
    const float4 v = *reinterpret_cast<const float4*>(
        &cube_s[(size_t)s * 4096 + r * 64 + c4 * 4]);
    *reinterpret_cast<float4*>(&sCs[r * LSTR + c4 * 4]) = v;
  }

  v8f tacc[4];
#pragma unroll
  for (int nt = 0; nt < 4; ++nt) tacc[nt] = v8f_zero();

  for (int x = 0; x < 64; ++x) {
    __syncthreads();  // protect Cr/Co/V from previous iteration's readers
#pragma unroll
    for (int it = 0; it < 8; ++it) {
      const int idx = tid + it * 128;
      const int r = idx >> 4, c4 = idx & 15;
      const float4 vr = *reinterpret_cast<const float4*>(
          &cube_r[(size_t)x * 4096 + r * 64 + c4 * 4]);
      *reinterpret_cast<float4*>(&sCr[r * LSTR + c4 * 4]) = vr;
      const float4 vo = *reinterpret_cast<const float4*>(
          &cube_o[(size_t)x * 4096 + r * 64 + c4 * 4]);
      // Co pair-interleaved for step-2 B reads: [y/2][o][2], y=r here.
      float* dst = &sCoP[(r >> 1) * PSTR + (r & 1)];
      dst[(c4 * 4 + 0) * 2] = vo.x;
      dst[(c4 * 4 + 1) * 2] = vo.y;
      dst[(c4 * 4 + 2) * 2] = vo.z;
      dst[(c4 * 4 + 3) * 2] = vo.w;
    }
    // Prefetch next x-slab toward L2/WGP$ early.
    if (x + 1 < 64 && tid < 64) {
      __builtin_prefetch(&cube_r[(size_t)(x + 1) * 4096 + tid * 64], 0, 3);
      __builtin_prefetch(&cube_o[(size_t)(x + 1) * 4096 + tid * 64], 0, 3);
    }
    __syncthreads();

    // Step 1: V[y,r]; wave owns y-rows [wave*16, wave*16+16).
    v8f vacc[4];
#pragma unroll
    for (int nt = 0; nt < 4; ++nt) vacc[nt] = v8f_zero();
#pragma unroll
    for (int st = 0; st < 16; ++st) {
      const int kb = st * 4 + hi * 2;  // k = z
      const v2f a = ld_v2f(&sCs[(wave * 16 + lm) * LSTR + kb]);
#pragma unroll
      for (int nt = 0; nt < 4; ++nt) {
        // B(k=z, n=r) = Cr[r,z]; (z,z+1) contiguous in sCr row.
        const v2f b = ld_v2f(&sCr[(nt * 16 + lm) * LSTR + kb]);
        vacc[nt] = wmma_f32(a, b, vacc[nt]);
      }
    }
    // Write V straight into step-2's A pair layout [y/2][r][2].
#pragma unroll
    for (int nt = 0; nt < 4; ++nt)
#pragma unroll
      for (int j = 0; j < 8; ++j) {
        const int y = wave * 16 + j + hi * 8;
        sVP[(y >> 1) * PSTR + (nt * 16 + lm) * 2 + (y & 1)] = vacc[nt][j];
      }
    __syncthreads();

    // Step 2: T_s[r,o] += V^T @ Co; wave owns r-rows [wave*16, wave*16+16).
#pragma unroll
    for (int st = 0; st < 16; ++st) {
      const int pr = st * 2 + hi;  // pair-row = (k=y)>>1
      const v2f a = ld_v2f(&sVP[pr * PSTR + (wave * 16 + lm) * 2]);
#pragma unroll
      for (int nt = 0; nt < 4; ++nt) {
        const v2f b = ld_v2f(&sCoP[pr * PSTR + (nt * 16 + lm) * 2]);
        tacc[nt] = wmma_f32(a, b, tacc[nt]);
      }
    }
  }
#pragma unroll
  for (int nt = 0; nt < 4; ++nt)
#pragma unroll
    for (int j = 0; j < 8; ++j) {
      const int r = wave * 16 + j + hi * 8;
      T[((size_t)s * 64 + r) * 64 + nt * 16 + lm] = tacc[nt][j];
    }
}

// ---------------------------------------------------------------------------
// out[b] = sum_{s,r} p1[b,s]*p2[b,r]*Q[b,sr],  Q tile = P3[16,64] @ T[sr,o]^T.
// 16 batch rows per WG; Q never hits memory: each 16x16 tile is weighted and
// reduced straight out of the accumulator; rows merge via LDS ds_add_f32.
// ---------------------------------------------------------------------------
__global__ __launch_bounds__(128) void triangle_kernel(
    const float* __restrict__ P1, const float* __restrict__ P2,
    const float* __restrict__ P3, const float* __restrict__ T,
    float* __restrict__ out) {
  __shared__ float sP1[16 * LSTR];
  __shared__ float sP2[16 * LSTR];
  __shared__ float sP3[16 * LSTR];
  __shared__ float sOut[16];
  const int tid = threadIdx.x, wave = tid >> 5, lane = tid & 31;
  const int lm = lane & 15, hi = lane >> 4;
  const int rowBase = blockIdx.x * 16;

  for (int idx = tid; idx < 16 * 64; idx += 128) {
    const int r = idx >> 6, c = idx & 63;
    sP1[r * LSTR + c] = P1[(size_t)(rowBase + r) * 64 + c];
    sP2[r * LSTR + c] = P2[(size_t)(rowBase + r) * 64 + c];
    sP3[r * LSTR + c] = P3[(size_t)(rowBase + r) * 64 + c];
  }
  if (tid < 16) sOut[tid] = 0.0f;
  __syncthreads();

  float partial[8];
#pragma unroll
  for (int j = 0; j < 8; ++j) partial[j] = 0.0f;

  // 256 sr-tiles of 16 columns; within a tile s is constant, r = r0 + col.
  for (int t = wave; t < 256; t += 4) {
    const int s = t >> 2;
    const int r0 = (t & 3) * 16;
    const float* Tb = &T[(size_t)t * 16 * 64];  // 16 sr-rows x 64 o, row-major
    v8f q = v8f_zero();
#pragma unroll
    for (int st = 0; st < 16; ++st) {
      const int kb = st * 4 + hi * 2;  // k = o
      const v2f a = ld_v2f(&sP3[lm * LSTR + kb]);  // A(m=batch row, k=o)
      const v2f b = ld_v2f(&Tb[lm * 64 + kb]);     // B(k=o, n=sr): global b64
      q = wmma_f32(a, b, q);
    }
#pragma unroll
    for (int j = 0; j < 8; ++j) {
      const int row = j + hi * 8;
      partial[j] += q[j] * sP1[row * LSTR + s] * sP2[row * LSTR + r0 + lm];
    }
  }
#pragma unroll
  for (int j = 0; j < 8; ++j) atomicAdd(&sOut[j + hi * 8], partial[j]);
  __syncthreads();
  if (tid < 16) out[rowBase + tid] = sOut[tid];
}

// ---------------------------------------------------------------------------
extern "C" void kernel_launch(void* const* d_in, const int* in_sizes, int n_in,
                              void* d_out, int out_size, void* d_ws,
                              size_t ws_size, hipStream_t stream) {
  const float* input1 = (const float*)d_in[0];
  const float* input2 = (const float*)d_in[1];
  const float* input3 = (const float*)d_in[2];
  const float* proj_s = (const float*)d_in[3];
  const float* proj_r = (const float*)d_in[4];
  const float* proj_o = (const float*)d_in[5];
  const float* cube_s = (const float*)d_in[6];
  const float* cube_r = (const float*)d_in[7];
  const float* cube_o = (const float*)d_in[8];
  const float* W21 = (const float*)d_in[9];
  const float* b21 = (const float*)d_in[10];
  const float* W22 = (const float*)d_in[11];
  const float* b22 = (const float*)d_in[12];
  const float* W23 = (const float*)d_in[13];
  const float* b23 = (const float*)d_in[14];
  float* out = (float*)d_out;

  const int B = 16384;
  const size_t BN = (size_t)B * 64;  // one [B,64] panel
  float* ws = (float*)d_ws;
  float* P1 = ws + 0 * BN;
  float* P2 = ws + 1 * BN;
  float* P3 = ws + 2 * BN;
  float* HA = ws + 3 * BN;  // MLP ping
  float* HB = ws + 4 * BN;  // MLP pong
  float* Tt = ws + 5 * BN;  // T[s,r,o], 64^3 floats (~22 MB total workspace)

  const dim3 blk(128);
  const float* nofp = nullptr;

  // Batch-independent core tensor (launched first; independent of GEMMs).
  hipLaunchKernelGGL(tcube_kernel, dim3(64), blk, 0, stream,
                     cube_s, cube_r, cube_o, Tt);
  // Projections + relation MLP.
  hipLaunchKernelGGL(fused_gemm_bx64, dim3(B / 64), blk, 0, stream,
                     input1, proj_s, nofp, P1, 512, 0, 0);
  hipLaunchKernelGGL(fused_gemm_bx64, dim3(B / 64), blk, 0, stream,
                     input3, proj_o, nofp, P3, 512, 0, 0);
  hipLaunchKernelGGL(fused_gemm_bx64, dim3(B / 64), blk, 0, stream,
                     input2, W21, b21, HA, 512, 1, 1);
  hipLaunchKernelGGL(fused_gemm_bx64, dim3(B / 64), blk, 0, stream,
                     HA, W22, b22, HB, 64, 1, 1);
  hipLaunchKernelGGL(fused_gemm_bx64, dim3(B / 64), blk, 0, stream,
                     HB, W23, b23, HA, 64, 1, 0);
  hipLaunchKernelGGL(fused_gemm_bx64, dim3(B / 64), blk, 0, stream,
                     HA, proj_r, nofp, P2, 64, 0, 0);
  // Fused Q-GEMM + trilinear reduction.
  hipLaunchKernelGGL(triangle_kernel, dim3(B / 16), blk, 0, stream,
                     P1, P2, P3, Tt, out);
}